// GConvGRUCell_27101243638400
// MI455X (gfx1250) — compile-verified
//
#include <hip/hip_runtime.h>
#include <hip/hip_bf16.h>
#include <math.h>

typedef __attribute__((ext_vector_type(2))) float v2f;
typedef __attribute__((ext_vector_type(8))) float v8f;

#define HID 64
#define CAT 128

// ---------------------------------------------------------------------------
// Degree count: one thread per edge, atomicAdd 1.0 into deg[dst]
// ---------------------------------------------------------------------------
__global__ __launch_bounds__(256) void deg_kernel(const int* __restrict__ dst,
                                                  float* __restrict__ deg, int E) {
    int e = blockIdx.x * blockDim.x + threadIdx.x;
    if (e < E) atomicAdd(&deg[dst[e]], 1.0f);
}

// ---------------------------------------------------------------------------
// Edge scatter: xfeat = [left(row,0..63) | right(row,0..63)] concatenated.
// One thread handles one 4-float chunk of one edge (32 chunks / edge).
// Chunks never straddle the 64-feature boundary, so a single float4 load works.
// ---------------------------------------------------------------------------
__global__ __launch_bounds__(256) void scatter_cat_kernel(
    const float* __restrict__ left,   // N x 64 (inputs)
    const float* __restrict__ right,  // N x 64 (state  OR  r*state)
    const int* __restrict__ src, const int* __restrict__ dst,
    float* __restrict__ agg,          // N x 128 accumulator
    int E)
{
    int t = blockIdx.x * blockDim.x + threadIdx.x;
    int total = E * 32;
    if (t >= total) return;
    int e = t >> 5;
    int f = (t & 31) * 4;           // feature offset 0..124, multiple of 4
    int s = src[e];
    int d = dst[e];
    const float* base = (f < HID) ? (left + s * HID + f)
                                  : (right + s * HID + (f - HID));
    float4 v = *(const float4*)base;
    float* o = agg + d * CAT + f;
    atomicAdd(o + 0, v.x);
    atomicAdd(o + 1, v.y);
    atomicAdd(o + 2, v.z);
    atomicAdd(o + 3, v.w);
}

// ---------------------------------------------------------------------------
// Gate GEMM: h = xs @ Wg_self + mean_agg @ Wg_neigh + bg + gate_bias
//            h = sigmoid(h); r = h[:, :64]; u = h[:, 64:]
// Writes: rstate = r * state, and u.
// Block = 256 threads = 8 waves; wave w owns output cols [16w, 16w+16).
// Each block owns node rows [16*blockIdx.x, +16).  K = 128, stepped by 4.
// ---------------------------------------------------------------------------
__global__ __launch_bounds__(256) void gate_gemm_kernel(
    const float* __restrict__ inputs, const float* __restrict__ state,
    const float* __restrict__ agg,    const float* __restrict__ deg,
    const float* __restrict__ Wg_self, const float* __restrict__ Wg_neigh,
    const float* __restrict__ bg, const float* __restrict__ gate_bias,
    float* __restrict__ u_out, float* __restrict__ rstate_out, int n_nodes)
{
    const int lane = threadIdx.x & 31;
    const int wave = threadIdx.x >> 5;
    const int half = lane >> 4;      // 0: K even pair base, 1: K odd pair base
    const int m    = lane & 15;      // M for A-frags, N for B-frags & C/D
    const int n0   = wave * 16;      // output column tile (8 tiles cover 128)

    int row = blockIdx.x * 16 + m;
    if (row >= n_nodes) row = n_nodes - 1;          // clamped load row (EXEC all-1)

    const float invdeg = 1.0f / fmaxf(deg[row], 1.0f);

    v8f acc = {};
#pragma unroll 4
    for (int k0 = 0; k0 < CAT; k0 += 4) {
        const int kk = k0 + 2 * half;               // kk, kk+1 in same 64-block
        // A (self): xs[row][kk..kk+1]
        const float* xsrc = (kk < HID) ? (inputs + row * HID + kk)
                                       : (state  + row * HID + kk - HID);
        v2f a1; a1.x = xsrc[0]; a1.y = xsrc[1];
        // A (neigh): mean-agg, scale folded into A rows
        const float* ag = agg + row * CAT + kk;
        v2f a2; a2.x = ag[0] * invdeg; a2.y = ag[1] * invdeg;
        // B frags: W[k][n], row-major with 128 cols
        const float* w1 = Wg_self  + kk * 128 + n0 + m;
        const float* w2 = Wg_neigh + kk * 128 + n0 + m;
        v2f b1; b1.x = w1[0]; b1.y = w1[128];
        v2f b2; b2.x = w2[0]; b2.y = w2[128];
        acc = __builtin_amdgcn_wmma_f32_16x16x4_f32(false, a1, false, b1,
                                                    (short)0, acc, false, false);
        acc = __builtin_amdgcn_wmma_f32_16x16x4_f32(false, a2, false, b2,
                                                    (short)0, acc, false, false);
    }

    const int n = n0 + m;                           // global output column
    const float bias = bg[n] + gate_bias[n];
#pragma unroll
    for (int v = 0; v < 8; v++) {
        const int rg = blockIdx.x * 16 + v + 8 * half;   // C/D layout row
        if (rg < n_nodes) {
            const float h = 1.0f / (1.0f + expf(-(acc[v] + bias)));
            if (n < HID) {
                rstate_out[rg * HID + n] = h * state[rg * HID + n];   // r * state
            } else {
                u_out[rg * HID + (n - HID)] = h;                      // u
            }
        }
    }
}

// ---------------------------------------------------------------------------
// Candidate GEMM: c = tanh(xc @ Wc_self + mean_agg2 @ Wc_neigh + bc + cand_bias)
// out = u * state + (1 - u) * c
// Block = 128 threads = 4 waves covering the 64 output columns.
// ---------------------------------------------------------------------------
__global__ __launch_bounds__(128) void cand_gemm_kernel(
    const float* __restrict__ inputs, const float* __restrict__ rstate,
    const float* __restrict__ state,
    const float* __restrict__ agg,   const float* __restrict__ deg,
    const float* __restrict__ Wc_self, const float* __restrict__ Wc_neigh,
    const float* __restrict__ bc, const float* __restrict__ cand_bias,
    const float* __restrict__ u_in, float* __restrict__ out, int n_nodes)
{
    const int lane = threadIdx.x & 31;
    const int wave = threadIdx.x >> 5;
    const int half = lane >> 4;
    const int m    = lane & 15;
    const int n0   = wave * 16;                    // 4 tiles cover 64 cols

    int row = blockIdx.x * 16 + m;
    if (row >= n_nodes) row = n_nodes - 1;

    const float invdeg = 1.0f / fmaxf(deg[row], 1.0f);

    v8f acc = {};
#pragma unroll 4
    for (int k0 = 0; k0 < CAT; k0 += 4) {
        const int kk = k0 + 2 * half;
        const float* xsrc = (kk < HID) ? (inputs + row * HID + kk)
                                       : (rstate + row * HID + kk - HID);
        v2f a1; a1.x = xsrc[0]; a1.y = xsrc[1];
        const float* ag = agg + row * CAT + kk;
        v2f a2; a2.x = ag[0] * invdeg; a2.y = ag[1] * invdeg;
        const float* w1 = Wc_self  + kk * HID + n0 + m;   // 64-col row-major
        const float* w2 = Wc_neigh + kk * HID + n0 + m;
        v2f b1; b1.x = w1[0]; b1.y = w1[HID];
        v2f b2; b2.x = w2[0]; b2.y = w2[HID];
        acc = __builtin_amdgcn_wmma_f32_16x16x4_f32(false, a1, false, b1,
                                                    (short)0, acc, false, false);
        acc = __builtin_amdgcn_wmma_f32_16x16x4_f32(false, a2, false, b2,
                                                    (short)0, acc, false, false);
    }

    const int n = n0 + m;
    const float bias = bc[n] + cand_bias[n];
#pragma unroll
    for (int v = 0; v < 8; v++) {
        const int rg = blockIdx.x * 16 + v + 8 * half;
        if (rg < n_nodes) {
            const float c = tanhf(acc[v] + bias);
            const float u = u_in[rg * HID + n];
            const float s = state[rg * HID + n];
            out[rg * HID + n] = u * s + (1.0f - u) * c;
        }
    }
}

// ---------------------------------------------------------------------------
// Host-side launch
// ---------------------------------------------------------------------------
extern "C" void kernel_launch(void* const* d_in, const int* in_sizes, int n_in,
                              void* d_out, int out_size, void* d_ws, size_t ws_size,
                              hipStream_t stream) {
    const float* inputs    = (const float*)d_in[0];
    const float* state     = (const float*)d_in[1];
    const int*   src       = (const int*)d_in[2];
    const int*   dst       = (const int*)d_in[3];
    const float* Wg_self   = (const float*)d_in[4];
    const float* Wg_neigh  = (const float*)d_in[5];
    const float* bg        = (const float*)d_in[6];
    const float* Wc_self   = (const float*)d_in[7];
    const float* Wc_neigh  = (const float*)d_in[8];
    const float* bc        = (const float*)d_in[9];
    const float* gate_bias = (const float*)d_in[10];
    const float* cand_bias = (const float*)d_in[11];
    float*       out       = (float*)d_out;

    const int N = in_sizes[0] / HID;      // 100000
    const int E = in_sizes[2];            // 1600000

    // Workspace layout (floats): deg[N] | agg[N*128] | u[N*64] | rstate[N*64]
    float* deg    = (float*)d_ws;
    float* agg    = deg + N;
    float* u_ws   = agg + (size_t)N * CAT;
    float* rstate = u_ws + (size_t)N * HID;

    const int node_tiles = (N + 15) / 16;
    const int scat_blocks = (E * 32 + 255) / 256;

    // Pass 1: degree + scatter xs = [inputs | state]
    hipMemsetAsync(deg, 0, (size_t)N * sizeof(float), stream);
    hipMemsetAsync(agg, 0, (size_t)N * CAT * sizeof(float), stream);
    deg_kernel<<<(E + 255) / 256, 256, 0, stream>>>(dst, deg, E);
    scatter_cat_kernel<<<scat_blocks, 256, 0, stream>>>(inputs, state, src, dst, agg, E);

    // Gate GEMM (sigmoid) -> u, r*state
    gate_gemm_kernel<<<node_tiles, 256, 0, stream>>>(
        inputs, state, agg, deg, Wg_self, Wg_neigh, bg, gate_bias,
        u_ws, rstate, N);

    // Pass 2: scatter xc = [inputs | r*state]
    hipMemsetAsync(agg, 0, (size_t)N * CAT * sizeof(float), stream);
    scatter_cat_kernel<<<scat_blocks, 256, 0, stream>>>(inputs, rstate, src, dst, agg, E);

    // Candidate GEMM (tanh) + GRU blend -> out
    cand_gemm_kernel<<<node_tiles, 128, 0, stream>>>(
        inputs, rstate, state, agg, deg, Wc_self, Wc_neigh, bc, cand_bias,
        u_ws, out, N);
}